// GCN_8332236554730
// MI455X (gfx1250) — compile-verified
//
#include <hip/hip_runtime.h>
#include <hip/hip_bf16.h>

// ---------------------------------------------------------------------------
// GCN: 3 x [SpMM (copy_src + segment_sum) -> Linear -> act] on gfx1250.
//   N = 50000 nodes, d = 300, E = 1.6M edges.
// SpMM: wave-per-edge gather + global fp32 atomic scatter (L2-resident, 60MB).
// GEMM: V_WMMA_F32_16X16X4_F32, wave computes 16x64 tile, K stepped by 4.
// Ragged N edge handled by CLAMPED (in-bounds) loads: garbage contributions
// stay confined to output columns >= 300 which are never stored, so the
// K-loop has no divergence and EXEC is trivially all-1s at every WMMA.
// ---------------------------------------------------------------------------

#define N_NODES 50000
#define DFEAT   300
#define MTILES  (N_NODES / 16)   // 3125 exact
#define NCOLG   5                // 5 * 64 = 320 >= 300 columns

typedef float v2f __attribute__((ext_vector_type(2)));
typedef float v8f __attribute__((ext_vector_type(8)));

// ---------------------------------------------------------------------------
// Zero-fill (agg buffer must start at 0 before atomic scatter).
// ---------------------------------------------------------------------------
__global__ void gcn_zero_f32(float* __restrict__ p, long n) {
    long i = (long)blockIdx.x * blockDim.x + threadIdx.x;
    long stride = (long)gridDim.x * blockDim.x;
    for (; i < n; i += stride) p[i] = 0.0f;
}

// ---------------------------------------------------------------------------
// SpMM scatter:  agg[dst[e], :] += x[src[e], :]
// One wave per edge; lanes stride the 300-float row (coalesced 128B reads,
// fp32 atomics land in L2 — both x and agg are 60MB, fully L2-resident).
// ---------------------------------------------------------------------------
__global__ void gcn_spmm_scatter(const float* __restrict__ x,
                                 const int*   __restrict__ src,
                                 const int*   __restrict__ dst,
                                 float*       __restrict__ agg,
                                 int E) {
    const int  lane  = threadIdx.x & 31;
    const long wid   = ((long)blockIdx.x * blockDim.x + threadIdx.x) >> 5;
    const long nwave = ((long)gridDim.x * blockDim.x) >> 5;
    for (long e = wid; e < E; e += nwave) {
        const int s = src[e];          // wave-uniform (scalar load)
        const int t = dst[e];
        const float* __restrict__ xs = x   + (long)s * DFEAT;
        float*       __restrict__ ot = agg + (long)t * DFEAT;
        #pragma unroll 1
        for (int j = lane; j < DFEAT; j += 32) {
            atomicAdd(ot + j, xs[j]);  // global_atomic_add_f32
        }
    }
}

// ---------------------------------------------------------------------------
// Dense GEMM + bias + activation via V_WMMA_F32_16X16X4_F32.
//   out[m, n] = act( sum_k A[m, k] * W[k, n] + b[n] )
// A: [50000 x 300]  W: [300 x 300]  b: [300]
//
// Per CDNA5 ISA VGPR layouts (f32, wave32):
//   A 16x4 : lane<16 -> {K=kb+0, kb+1}, lane>=16 -> {K=kb+2, kb+3}, M=lane%16
//   B 4x16 : v.x -> row K=kb+halfk,   v.y -> row K=kb+halfk+1,   N=lane%16
//   C/D    : VGPR r -> M = m0 + r + (lane>=16 ? 8 : 0), N = n0 + lane%16
// ---------------------------------------------------------------------------
__device__ __forceinline__ v8f wmma_f32_16x16x4(v2f a, v2f b, v8f c) {
    // 8 args: (neg_a, A, neg_b, B, c_mod, C, reuse_a, reuse_b)
    return __builtin_amdgcn_wmma_f32_16x16x4_f32(
        false, a, false, b, (short)0, c, false, false);
}

template <int ACT> // 0 = ReLU, 1 = sigmoid
__global__ void gcn_gemm_bias_act(const float* __restrict__ A,
                                  const float* __restrict__ W,
                                  const float* __restrict__ bias,
                                  float*       __restrict__ out) {
    const int  lane  = threadIdx.x & 31;
    const int  ln    = lane & 15;
    const int  hi    = lane >> 4;        // 0 for lanes 0-15, 1 for 16-31
    const int  halfk = hi << 1;          // 0 or 2
    const long wave  = (long)blockIdx.x * (blockDim.x >> 5) + (threadIdx.x >> 5);

    const long totalWaves = (long)MTILES * NCOLG;
    if (wave >= totalWaves) return;      // wave-uniform exit (last block only)

    const int m0 = (int)(wave / NCOLG) * 16;
    const int n0 = (int)(wave % NCOLG) * 64;

    // Per-lane column index for each of the 4 sub-tiles, clamped in-bounds.
    // Clamping (not zeroing) is sufficient: a wrong value in B column n only
    // affects output column n, and columns >= 300 are never stored.
    const int nc0 = min(n0 +  0 + ln, DFEAT - 1);
    const int nc1 = min(n0 + 16 + ln, DFEAT - 1);
    const int nc2 = min(n0 + 32 + ln, DFEAT - 1);
    const int nc3 = min(n0 + 48 + ln, DFEAT - 1);

    v8f acc0 = {}, acc1 = {}, acc2 = {}, acc3 = {};

    const float* __restrict__ arow = A + (long)(m0 + ln) * DFEAT + halfk;
    const float* __restrict__ wrow = W + (long)halfk * DFEAT;  // advances by 4 rows/step

    for (int kb = 0; kb < DFEAT; kb += 4) {
        v2f a;
        a.x = arow[kb];
        a.y = arow[kb + 1];

        const float* __restrict__ w0 = wrow + (long)kb * DFEAT; // row kb+halfk
        const float* __restrict__ w1 = w0 + DFEAT;              // row kb+halfk+1

        v2f bf0; bf0.x = w0[nc0]; bf0.y = w1[nc0];
        acc0 = wmma_f32_16x16x4(a, bf0, acc0);
        v2f bf1; bf1.x = w0[nc1]; bf1.y = w1[nc1];
        acc1 = wmma_f32_16x16x4(a, bf1, acc1);
        v2f bf2; bf2.x = w0[nc2]; bf2.y = w1[nc2];
        acc2 = wmma_f32_16x16x4(a, bf2, acc2);
        v2f bf3; bf3.x = w0[nc3]; bf3.y = w1[nc3];
        acc3 = wmma_f32_16x16x4(a, bf3, acc3);
    }

    // Store: D VGPR r -> row m0 + r + 8*hi, col n0 + s*16 + ln.
    const int mrow = m0 + (hi << 3);
    #pragma unroll
    for (int s = 0; s < 4; ++s) {
        const int n = n0 + s * 16 + ln;
        if (n >= DFEAT) continue;        // plain store masking (no WMMA after)
        const float bn = bias[n];
        const v8f acc = (s == 0) ? acc0 : (s == 1) ? acc1 : (s == 2) ? acc2 : acc3;
        #pragma unroll
        for (int r = 0; r < 8; ++r) {
            float v = acc[r] + bn;
            if (ACT == 0) v = v > 0.0f ? v : 0.0f;
            else          v = 1.0f / (1.0f + __expf(-v));
            out[(long)(mrow + r) * DFEAT + n] = v;
        }
    }
}

// ---------------------------------------------------------------------------
// Host-side orchestration (graph-capture safe: launches only, all on stream).
// d_ws layout: [bufA: 60MB agg][bufB: 60MB activations]
// ---------------------------------------------------------------------------
extern "C" void kernel_launch(void* const* d_in, const int* in_sizes, int n_in,
                              void* d_out, int out_size, void* d_ws, size_t ws_size,
                              hipStream_t stream) {
    const float* feat = (const float*)d_in[0];
    const int*   src  = (const int*)  d_in[1];
    const int*   dst  = (const int*)  d_in[2];
    const float* W1   = (const float*)d_in[3];
    const float* b1   = (const float*)d_in[4];
    const float* W2   = (const float*)d_in[5];
    const float* b2   = (const float*)d_in[6];
    const float* W3   = (const float*)d_in[7];
    const float* b3   = (const float*)d_in[8];
    float* out = (float*)d_out;

    const int  E = in_sizes[1];                 // 1.6M edges
    const long n = (long)N_NODES * DFEAT;       // 15M floats per buffer

    float* bufA = (float*)d_ws;                 // aggregation target
    float* bufB = bufA + n;                     // layer activations

    const dim3 blk(256);
    const int zeroBlocks = 2048;
    const int scatBlocks = (E + 7) / 8;                      // 8 waves/block, 1 edge/wave
    const int gemmBlocks = (int)(((long)MTILES * NCOLG + 7) / 8);

    // ---- Layer 1: agg = SpMM(features); x1 = relu(agg @ W1 + b1) ----
    gcn_zero_f32<<<zeroBlocks, blk, 0, stream>>>(bufA, n);
    gcn_spmm_scatter<<<scatBlocks, blk, 0, stream>>>(feat, src, dst, bufA, E);
    gcn_gemm_bias_act<0><<<gemmBlocks, blk, 0, stream>>>(bufA, W1, b1, bufB);

    // ---- Layer 2: agg = SpMM(x1); x2 = relu(agg @ W2 + b2) ----
    gcn_zero_f32<<<zeroBlocks, blk, 0, stream>>>(bufA, n);
    gcn_spmm_scatter<<<scatBlocks, blk, 0, stream>>>(bufB, src, dst, bufA, E);
    gcn_gemm_bias_act<0><<<gemmBlocks, blk, 0, stream>>>(bufA, W2, b2, bufB);

    // ---- Layer 3: agg = SpMM(x2); out = sigmoid(agg @ W3 + b3) ----
    gcn_zero_f32<<<zeroBlocks, blk, 0, stream>>>(bufA, n);
    gcn_spmm_scatter<<<scatBlocks, blk, 0, stream>>>(bufB, src, dst, bufA, E);
    gcn_gemm_bias_act<1><<<gemmBlocks, blk, 0, stream>>>(bufA, W3, b3, out);
}